// GNNLayer_16252156248657
// MI455X (gfx1250) — compile-verified
//
#include <hip/hip_runtime.h>
#include <hip/hip_bf16.h>
#include <math.h>

typedef __attribute__((ext_vector_type(2))) float v2f;
typedef __attribute__((ext_vector_type(8))) float v8f;

#define KDIM 256
#define NDIM 64

// ---------------------------------------------------------------------------
// Zero the output accumulator (d_out is poisoned by the harness; atomics need
// a clean accumulator every call).
// ---------------------------------------------------------------------------
__global__ void zero_out_kernel(float4* __restrict__ out, int n4) {
    int i = blockIdx.x * blockDim.x + threadIdx.x;
    if (i < n4) out[i] = make_float4(0.f, 0.f, 0.f, 0.f);
}

// ---------------------------------------------------------------------------
// support = features @ W  via V_WMMA_F32_16X16X4_F32 (exact fp32 path).
// One wave per 16-row M tile; each wave produces the full 16x64 tile,
// reusing the A fragment across the 4 N tiles.
//
// A frag (16x4 f32, 2 VGPR): lane L -> row = L&15, VGPR0 = A[row][k0+2h],
//   VGPR1 = A[row][k0+2h+1], h = L>>4   (contiguous float2 load)
// B frag (4x16 f32, 2 VGPR): lane L -> col = L&15, VGPR0 = B[k0+2h][col],
//   VGPR1 = B[k0+2h+1][col]
// C/D (16x16 f32, 8 VGPR): element v -> M = v + 8h, N = col
// ---------------------------------------------------------------------------
__global__ void __launch_bounds__(256)
gemm_wmma_kernel(const float* __restrict__ feat,   // [M,256] row-major
                 const float* __restrict__ wt,     // [256,64] row-major
                 float* __restrict__ support,      // [M,64]
                 int mtiles) {
    const int wave  = (int)(blockIdx.x * (blockDim.x >> 5) + (threadIdx.x >> 5));
    if (wave >= mtiles) return;               // uniform per wave: EXEC all-ones inside

    const int lane = threadIdx.x & 31;
    const int row  = lane & 15;               // M index within tile (A) / N col (B,C,D)
    const int half = lane >> 4;
    const int kk   = half * 2;

    const size_t m0 = (size_t)wave * 16;
    const float* arow = feat + (m0 + (size_t)row) * KDIM;

    v8f acc0 = {}, acc1 = {}, acc2 = {}, acc3 = {};

    #pragma unroll 4
    for (int k0 = 0; k0 < KDIM; k0 += 4) {
        // A fragment: contiguous 8B load
        v2f a;
        a.x = arow[k0 + kk + 0];
        a.y = arow[k0 + kk + 1];

        const float* w0 = wt + (size_t)(k0 + kk) * NDIM + row;      // K = k0+kk
        const float* w1 = w0 + NDIM;                                // K = k0+kk+1

        v2f b0, b1, b2, b3;
        b0.x = w0[0];   b0.y = w1[0];
        b1.x = w0[16];  b1.y = w1[16];
        b2.x = w0[32];  b2.y = w1[32];
        b3.x = w0[48];  b3.y = w1[48];

        acc0 = __builtin_amdgcn_wmma_f32_16x16x4_f32(false, a, false, b0, (short)0, acc0, false, false);
        acc1 = __builtin_amdgcn_wmma_f32_16x16x4_f32(false, a, false, b1, (short)0, acc1, false, false);
        acc2 = __builtin_amdgcn_wmma_f32_16x16x4_f32(false, a, false, b2, (short)0, acc2, false, false);
        acc3 = __builtin_amdgcn_wmma_f32_16x16x4_f32(false, a, false, b3, (short)0, acc3, false, false);
    }

    // Store 16x64 tile. Element v of acc -> M = m0 + v + 8*half, N = jt*16 + row.
    float* orow = support + (m0 + (size_t)half * 8) * NDIM + row;
    #pragma unroll
    for (int v = 0; v < 8; ++v) {
        float* p = orow + (size_t)v * NDIM;
        p[0]  = acc0[v];
        p[16] = acc1[v];
        p[32] = acc2[v];
        p[48] = acc3[v];
    }
}

// ---------------------------------------------------------------------------
// COO SpMM scatter: one wave32 per edge, 2 columns per lane.
// support / out are L2-resident (12.8 MB each vs 192 MB L2), so the random
// gather + f32 atomic scatter run at L2 bandwidth.
// ---------------------------------------------------------------------------
__global__ void __launch_bounds__(256)
edge_scatter_kernel(const float* __restrict__ support,
                    const int* __restrict__ erow,
                    const int* __restrict__ ecol,
                    const float* __restrict__ ew,
                    float* __restrict__ out,
                    int E) {
    const int e    = (int)(blockIdx.x * (blockDim.x >> 5) + (threadIdx.x >> 5));
    const int lane = threadIdx.x & 31;
    if (e >= E) return;

    const int   c = ecol[e];
    const int   r = erow[e];
    const float w = ew[e];

    const float2* src = (const float2*)(support + (size_t)c * NDIM) + lane;
    float2 v = *src;

    float* dst = out + (size_t)r * NDIM + lane * 2;
    __hip_atomic_fetch_add(dst + 0, v.x * w, __ATOMIC_RELAXED, __HIP_MEMORY_SCOPE_AGENT);
    __hip_atomic_fetch_add(dst + 1, v.y * w, __ATOMIC_RELAXED, __HIP_MEMORY_SCOPE_AGENT);
}

// ---------------------------------------------------------------------------
// Epilogue: tanh if *active (device-side flag read, keeps launch deterministic)
// ---------------------------------------------------------------------------
__global__ void tanh_kernel(float* __restrict__ out, const int* __restrict__ active, int n) {
    int i = blockIdx.x * blockDim.x + threadIdx.x;
    if (i < n) {
        float v = out[i];
        out[i] = (*active) ? tanhf(v) : v;
    }
}

extern "C" void kernel_launch(void* const* d_in, const int* in_sizes, int n_in,
                              void* d_out, int out_size, void* d_ws, size_t ws_size,
                              hipStream_t stream) {
    const float* feat   = (const float*)d_in[0];   // [N,256]
    const float* wt     = (const float*)d_in[1];   // [256,64]
    const int*   erow   = (const int*)  d_in[2];   // [E]
    const int*   ecol   = (const int*)  d_in[3];   // [E]
    const float* ew     = (const float*)d_in[4];   // [E]
    const int*   active = (const int*)  d_in[5];   // scalar

    float* out     = (float*)d_out;                // [N,64]
    float* support = (float*)d_ws;                 // [N,64] scratch

    const int N      = in_sizes[0] / KDIM;         // 50000
    const int E      = in_sizes[2];                // 1,600,000
    const int mtiles = N / 16;                     // 3125 (N divisible by 16)

    // 1) zero accumulator
    {
        int n4 = out_size / 4;
        int blk = 256, grd = (n4 + blk - 1) / blk;
        zero_out_kernel<<<grd, blk, 0, stream>>>((float4*)out, n4);
    }
    // 2) dense projection via fp32 WMMA (8 waves / 256-thread block)
    {
        int wavesPerBlk = 8;
        int grd = (mtiles + wavesPerBlk - 1) / wavesPerBlk;
        gemm_wmma_kernel<<<grd, 256, 0, stream>>>(feat, wt, support, mtiles);
    }
    // 3) edge gather/scale/scatter-add (wave per edge)
    {
        int wavesPerBlk = 8;
        int grd = (E + wavesPerBlk - 1) / wavesPerBlk;
        edge_scatter_kernel<<<grd, 256, 0, stream>>>(support, erow, ecol, ew, out, E);
    }
    // 4) tanh epilogue
    {
        int blk = 256, grd = (out_size + blk - 1) / blk;
        tanh_kernel<<<grd, blk, 0, stream>>>(out, active, out_size);
    }
}